// VectorQuantizer_62405874811765
// MI455X (gfx1250) — compile-verified
//
#include <hip/hip_runtime.h>
#include <hip/hip_bf16.h>

typedef float v2f __attribute__((ext_vector_type(2)));
typedef float v4f __attribute__((ext_vector_type(4)));
typedef float v8f __attribute__((ext_vector_type(8)));
typedef int   v4i __attribute__((ext_vector_type(4)));

typedef __attribute__((address_space(1))) v4i gv4i;   // global int4
typedef __attribute__((address_space(3))) v4i lv4i;   // LDS int4

#define N_ROWS   32768          // 8*64*64
#define E_DIM    256
#define N_E      8192
#define ZTOT     (N_ROWS * E_DIM)   // 8388608
#define BETA     0.25f
#define TILE_CODES 16           // codes per pipeline stage
#define N_TILES   (N_E / TILE_CODES)
#define LDS_PITCH  260          // 256 + 4 pad floats -> conflict-free b64 LDS reads

#if defined(__has_builtin)
#  if __has_builtin(__builtin_amdgcn_global_load_async_to_lds_b128)
#    define VQ_ASYNC 1
#  endif
#endif

// ---------------------------------------------------------------------------
// Kernel 1: per-code squared norms ||e_k||^2 -> ws; also zero the loss accum.
// ---------------------------------------------------------------------------
__global__ __launch_bounds__(256, 1)
void vq_enorm(const float* __restrict__ emb,
              float* __restrict__ enorm,
              float* __restrict__ gsum) {
  const int lane = threadIdx.x & 31;
  const int wave = threadIdx.x >> 5;
  const int code = blockIdx.x * 8 + wave;

  const float* er = emb + (size_t)code * E_DIM + lane * 8;
  v4f a = *(const v4f*)(er);
  v4f b = *(const v4f*)(er + 4);
  float s = a.x*a.x + a.y*a.y + a.z*a.z + a.w*a.w
          + b.x*b.x + b.y*b.y + b.z*b.z + b.w*b.w;
  #pragma unroll
  for (int off = 16; off >= 1; off >>= 1) s += __shfl_xor(s, off);
  if (lane == 0) enorm[code] = s;
  if (blockIdx.x == 0 && threadIdx.x == 0) *gsum = 0.0f;
}

// ---------------------------------------------------------------------------
// Kernel 2: main pass. 8 waves/block, 16 z-rows per wave (A fragments resident
// in 128 VGPRs, reused across all 8192 codes). Codebook streamed through a
// double-buffered LDS pipeline (16 codes/stage): iteration t computes 64
// chained v_wmma_f32_16x16x4_f32 from buf[t&1] while tile t+1 streams into
// buf[t&1 ^ 1] via global_load_async_to_lds_b128 (ASYNCcnt-tracked). One
// barrier per stage.
// ---------------------------------------------------------------------------
__global__ __launch_bounds__(256, 1)
void vq_main(const float* __restrict__ z,
             const float* __restrict__ emb,
             const float* __restrict__ enorm,
             float* __restrict__ gsum,
             float* __restrict__ out_zq,
             float* __restrict__ out_idx) {
  __shared__ float lds[2][TILE_CODES * LDS_PITCH];
  __shared__ float lnorm[2][TILE_CODES];

  const int tid  = threadIdx.x;
  const int lane = tid & 31;
  const int wave = tid >> 5;
  const int row_base = blockIdx.x * 128 + wave * 16;

  // WMMA f32 16x16x4 fragment mapping: lanes 0-15 -> M/N=lane, K={0,1};
  // lanes 16-31 -> M/N=lane-16, K={2,3}.
  const int m  = lane & 15;
  const int kb = (lane >> 4) * 2;

  // Preload this wave's 16 z-rows as A fragments (64 x v2f = 128 VGPRs).
  const float* zr = z + (size_t)(row_base + m) * E_DIM;
  v2f areg[64];
  #pragma unroll
  for (int kk = 0; kk < 64; ++kk)
    areg[kk] = *(const v2f*)(zr + kk * 4 + kb);

  float minv[8];
  int   midx[8];
  #pragma unroll
  for (int j = 0; j < 8; ++j) { minv[j] = 3.4e38f; midx[j] = 0; }

  // Loader mapping: 16 threads per codebook row, 16 floats (4 x b128) each.
  const int ldr = tid >> 4;          // 0..15: code row within tile
  const int seg = (tid & 15) * 16;   // column segment start

  auto load_tile = [&](int tile, int buf) {
    const float* src = emb + (size_t)(tile + ldr) * E_DIM + seg;
    float*       dst = &lds[buf][ldr * LDS_PITCH + seg];
#if defined(VQ_ASYNC)
    #pragma unroll
    for (int q = 0; q < 4; ++q)
      __builtin_amdgcn_global_load_async_to_lds_b128(
          (gv4i*)(src + q * 4), (lv4i*)(dst + q * 4), 0, 0);
#else
    #pragma unroll
    for (int q = 0; q < 4; ++q)
      *(v4f*)(dst + q * 4) = *(const v4f*)(src + q * 4);
#endif
    if (tid < TILE_CODES) lnorm[buf][tid] = enorm[tile + tid];
  };

  auto stage_sync = [&]() {
#if defined(VQ_ASYNC)
#  if __has_builtin(__builtin_amdgcn_s_wait_asynccnt)
    __builtin_amdgcn_s_wait_asynccnt(0);
#  else
    asm volatile("s_wait_asynccnt 0x0" ::: "memory");
#  endif
#endif
    __syncthreads();
  };

  // Prologue: stage tile 0 into buffer 0.
  load_tile(0, 0);
  stage_sync();

  for (int it = 0; it < N_TILES; ++it) {
    const int buf = it & 1;
    if (it + 1 < N_TILES) load_tile((it + 1) * TILE_CODES, buf ^ 1);

    const float* bp = &lds[buf][m * LDS_PITCH + kb];
    v8f c = {};
    #pragma unroll
    for (int kk = 0; kk < 64; ++kk) {
      v2f b = *(const v2f*)(bp + kk * 4);
      c = __builtin_amdgcn_wmma_f32_16x16x4_f32(
              false, areg[kk], false, b, (short)0, c, false, false);
    }
    const int   code = it * TILE_CODES + m;   // this lane's code column
    const float en   = lnorm[buf][m];
    #pragma unroll
    for (int j = 0; j < 8; ++j) {
      float s = en - 2.0f * c[j];
      if (s < minv[j]) { minv[j] = s; midx[j] = code; }  // ascending codes
    }
    stage_sync();
  }

  // Per-row argmin across the 16 code columns (xor masks 1..8 stay inside each
  // 16-lane half; lower half = rows j, upper half = rows j+8). Tie -> min idx.
  #pragma unroll
  for (int j = 0; j < 8; ++j) {
    float v = minv[j];
    int   i = midx[j];
    #pragma unroll
    for (int off = 8; off >= 1; off >>= 1) {
      float ov = __shfl_xor(v, off);
      int   oi = __shfl_xor(i, off);
      if (ov < v || (ov == v && oi < i)) { v = ov; i = oi; }
    }
    midx[j] = i;   // every lane in a half now holds that half's row result
  }

  // idx output (as float, per harness output dtype)
  #pragma unroll
  for (int j = 0; j < 8; ++j) {
    if (lane == 0)  out_idx[row_base + j]     = (float)midx[j];
    if (lane == 16) out_idx[row_base + 8 + j] = (float)midx[j];
  }

  // Gather z_q = emb[idx] (forward value of z_q_ste) + MSE partial sum.
  float lsum = 0.0f;
  const int d0 = lane * 8;
  #pragma unroll
  for (int j = 0; j < 8; ++j) {
    const int clo = __shfl(midx[j], 0);    // row row_base+j
    const int chi = __shfl(midx[j], 16);   // row row_base+8+j
    {
      const float* e  = emb + (size_t)clo * E_DIM + d0;
      const float* zp = z   + (size_t)(row_base + j) * E_DIM + d0;
      float*       op = out_zq + (size_t)(row_base + j) * E_DIM + d0;
      v4f e0 = *(const v4f*)e,  e1 = *(const v4f*)(e + 4);
      v4f z0 = *(const v4f*)zp, z1 = *(const v4f*)(zp + 4);
      *(v4f*)op = e0; *(v4f*)(op + 4) = e1;
      v4f a = e0 - z0, b = e1 - z1;
      lsum += a.x*a.x + a.y*a.y + a.z*a.z + a.w*a.w
            + b.x*b.x + b.y*b.y + b.z*b.z + b.w*b.w;
    }
    {
      const float* e  = emb + (size_t)chi * E_DIM + d0;
      const float* zp = z   + (size_t)(row_base + 8 + j) * E_DIM + d0;
      float*       op = out_zq + (size_t)(row_base + 8 + j) * E_DIM + d0;
      v4f e0 = *(const v4f*)e,  e1 = *(const v4f*)(e + 4);
      v4f z0 = *(const v4f*)zp, z1 = *(const v4f*)(zp + 4);
      *(v4f*)op = e0; *(v4f*)(op + 4) = e1;
      v4f a = e0 - z0, b = e1 - z1;
      lsum += a.x*a.x + a.y*a.y + a.z*a.z + a.w*a.w
            + b.x*b.x + b.y*b.y + b.z*b.z + b.w*b.w;
    }
  }
  #pragma unroll
  for (int off = 16; off >= 1; off >>= 1) lsum += __shfl_xor(lsum, off);
  if (lane == 0) atomicAdd(gsum, lsum);
}

// ---------------------------------------------------------------------------
// Kernel 3: finalize scalar losses.
// ---------------------------------------------------------------------------
__global__ void vq_finalize(const float* __restrict__ gsum,
                            float* __restrict__ out) {
  const float mse = *gsum / (float)ZTOT;
  out[ZTOT]     = BETA * mse;   // loss_commit
  out[ZTOT + 1] = mse;          // loss_codebook
}

extern "C" void kernel_launch(void* const* d_in, const int* in_sizes, int n_in,
                              void* d_out, int out_size, void* d_ws, size_t ws_size,
                              hipStream_t stream) {
  (void)in_sizes; (void)n_in; (void)out_size; (void)ws_size;
  const float* z   = (const float*)d_in[0];   // [8,64,64,256] fp32
  const float* emb = (const float*)d_in[1];   // [8192,256] fp32

  float* out     = (float*)d_out;
  float* out_zq  = out;                       // ZTOT floats
  float* out_idx = out + ZTOT + 2;            // N_ROWS floats (after 2 scalars)

  float* enorm = (float*)d_ws;                // N_E floats
  float* gsum  = enorm + N_E;                 // 1 float

  vq_enorm<<<N_E / 8, 256, 0, stream>>>(emb, enorm, gsum);
  vq_main<<<N_ROWS / 128, 256, 0, stream>>>(z, emb, enorm, gsum, out_zq, out_idx);
  vq_finalize<<<1, 1, 0, stream>>>(gsum, out);
}